// AdditiveDTMGP_85298050499159
// MI455X (gfx1250) — compile-verified
//
#include <hip/hip_runtime.h>
#include <cstdint>
#include <cstddef>

#define MPTS 63

typedef __attribute__((ext_vector_type(2)))  float v2f;
typedef __attribute__((ext_vector_type(8)))  float v8f;

__device__ __forceinline__ float softplus_f(float x) {
  return log1pf(__expf(x));
}

// ---------------------------------------------------------------------------
// ws[0] = 0  (KL accumulator; re-zeroed every launch for graph replay)
// ---------------------------------------------------------------------------
__global__ void zero_kl_kernel(float* p) { p[0] = 0.0f; }

// ---------------------------------------------------------------------------
// Permuted combined weights (m-major K ordering, k = m*F + f):
//   A[(m*F+f)*16 + o] = sum_n Rinv[m,n] * (mu_w[f,o,n] + softplus(rho)*eps)
// Columns o >= O are zero-filled (stage-3 N padding).
// ---------------------------------------------------------------------------
__global__ void build_A_kernel(const float* __restrict__ Rinv,
                               const float* __restrict__ mu_w,
                               const float* __restrict__ rho_w,
                               const float* __restrict__ eps_w,
                               float* __restrict__ A, int F, int O) {
  const int idx = blockIdx.x * blockDim.x + threadIdx.x;
  const int tot = F * MPTS * 16;
  if (idx >= tot) return;
  const int o = idx & 15;
  const int k = idx >> 4;       // permuted K index: k = m*F + f
  const int f = k % F;          // F is 16 or 32 (prep kernel; cost irrelevant)
  const int m = k / F;
  float acc = 0.0f;
  if (o < O) {
    const size_t base = ((size_t)f * O + o) * MPTS;
    const float* mw = mu_w  + base;
    const float* rw = rho_w + base;
    const float* ew = eps_w + base;
    const float* rv = Rinv + (size_t)m * MPTS;
    for (int n = 0; n < MPTS; ++n) {
      const float s = softplus_f(rw[n]);
      acc += rv[n] * (mw[n] + s * ew[n]);
    }
  }
  A[idx] = acc;
}

// bias[o] = sum_f (mu_b[f,o] + softplus(rho_b)*eps_b);  o>=O -> 0
__global__ void build_bias_kernel(const float* __restrict__ mu_b,
                                  const float* __restrict__ rho_b,
                                  const float* __restrict__ eps_b,
                                  float* __restrict__ bias, int F, int O) {
  const int o = threadIdx.x;  // blockDim.x == 16
  float acc = 0.0f;
  if (o < O) {
    for (int f = 0; f < F; ++f) {
      const int i = f * O + o;
      acc += mu_b[i] + softplus_f(rho_b[i]) * eps_b[i];
    }
  }
  bias[o] = acc;
}

// KL( N(mu, softplus(rho)) || N(0,1) ) summed; LDS tree + one atomic per block
__global__ void kl_accum_kernel(const float* __restrict__ mu,
                                const float* __restrict__ rho,
                                int n, float* __restrict__ acc) {
  __shared__ float red[256];
  const int idx = blockIdx.x * 256 + threadIdx.x;
  float v = 0.0f;
  if (idx < n) {
    const float m = mu[idx];
    const float s = softplus_f(rho[idx]);
    v = 0.5f * (s * s + m * m) - __logf(s) - 0.5f;
  }
  red[threadIdx.x] = v;
  __syncthreads();
  for (int off = 128; off > 0; off >>= 1) {
    if (threadIdx.x < off) red[threadIdx.x] += red[threadIdx.x + off];
    __syncthreads();
  }
  if (threadIdx.x == 0) atomicAdd(acc, red[0]);
}

__global__ void write_scalar_kernel(const float* src, float* dst) { dst[0] = src[0]; }

// E[idx] = (exp(v), exp(-v))  -- packed so exp(-|h-u|) = min(E.x*EU.x, E.y*EU.y)
__global__ void exp_pack_kernel(const float* __restrict__ in,
                                float2* __restrict__ out, int n) {
  const int idx = blockIdx.x * blockDim.x + threadIdx.x;
  if (idx < n) {
    const float v = in[idx];
    out[idx] = make_float2(__expf(v), __expf(-v));
  }
}

// EU[m] = (exp(-U[m]), exp(U[m]))
__global__ void build_eu_kernel(const float* __restrict__ U,
                                float2* __restrict__ EU) {
  const int m = threadIdx.x;
  if (m < MPTS) {
    const float u = U[m];
    EU[m] = make_float2(__expf(-u), __expf(u));
  }
}

// ---------------------------------------------------------------------------
// Fused stage GEMM over permuted K (k = m*F + f, F power of two):
//   out[b,o] = sum_k min(E[b,f].x*EU[m].x, E[b,f].y*EU[m].y) * A[k,o] + bias[o]
// One wave per 16-row tile, v_wmma_f32_16x16x4_f32, Kdim = F*63 compile-time.
// Per K-step per lane: 1x b128 (E pair) + 1x b64 (EU) + 2x b32 (A) loads,
// 2x v_pk_mul + 2x v_min, 1 WMMA.  f/m from k via AND/SHR only.
// MODE 0: epilogue packs (exp(v),exp(-v)) into outE [B,16] for the next stage
// MODE 1: write column 0 raw to outRaw[b] (stage 3)
// ---------------------------------------------------------------------------
template <int F, int MODE>
__global__ __launch_bounds__(256) void tmgp_stage_wmma_kernel(
    const float2* __restrict__ E, const float2* __restrict__ EU,
    const float* __restrict__ A, const float* __restrict__ bias,
    float2* __restrict__ outE, float* __restrict__ outRaw) {
  constexpr int Kdim  = F * MPTS;
  constexpr int LOG2F = (F == 32) ? 5 : 4;
  const int lane = threadIdx.x & 31;
  const int wave = threadIdx.x >> 5;
  const int row  = lane & 15;   // A: M index / B,C,D: N index
  const int half = lane >> 4;
  const int b0   = (blockIdx.x * 8 + wave) * 16;
  // E row for this lane's A-matrix row, viewed as float4 = two packed float2
  const float4* __restrict__ Erow =
      (const float4*)(E + (size_t)(b0 + row) * F);
  v8f acc = {};

#if defined(__HIP_DEVICE_COMPILE__)
  // ISA 16x4 f32 A layout: lanes 0-15 hold {K=0,K=1}; lanes 16-31 hold {K=2,K=3}
#pragma unroll 4
  for (int kk = 0; kk < Kdim; kk += 4) {
    const int k0 = kk + 2 * half;          // even; k0,k0+1 share the same m
    const int f0 = k0 & (F - 1);
    const int m0 = k0 >> LOG2F;
    const float4 e = Erow[f0 >> 1];        // (e^h_f0, e^-h_f0, e^h_f0+1, e^-h_f0+1)
    const float2 u = EU[m0];               // (e^-U_m, e^U_m)
    v2f a, b;
    a.x = fminf(e.x * u.x, e.y * u.y);     // exp(-|h_f0   - U_m|)
    a.y = fminf(e.z * u.x, e.w * u.y);     // exp(-|h_f0+1 - U_m|)
    b.x = A[(size_t)k0 * 16 + row];
    b.y = A[(size_t)k0 * 16 + 16 + row];
    acc = __builtin_amdgcn_wmma_f32_16x16x4_f32(false, a, false, b, (short)0,
                                                acc, false, false);
  }
#else
  (void)E; (void)EU; (void)A; (void)Erow;  // host-pass semantic stub
#endif

  const float bv = bias[row];
  // C/D layout: VGPR j -> output row b0 + j + 8*half, column = row
  if (MODE == 0) {
#pragma unroll
    for (int j = 0; j < 8; ++j) {
      const int br = b0 + j + 8 * half;
      const float v = acc[j] + bv;
      outE[(size_t)br * 16 + row] = make_float2(__expf(v), __expf(-v));
    }
  } else {
    if (row == 0) {
#pragma unroll
      for (int j = 0; j < 8; ++j) {
        const int br = b0 + j + 8 * half;
        outRaw[br] = acc[j] + bv;
      }
    }
  }
}

// ---------------------------------------------------------------------------
extern "C" void kernel_launch(void* const* d_in, const int* in_sizes, int n_in,
                              void* d_out, int out_size, void* d_ws, size_t ws_size,
                              hipStream_t stream) {
  (void)n_in; (void)out_size; (void)ws_size;
  const float* x      = (const float*)d_in[0];
  const float* U      = (const float*)d_in[1];
  const float* Rinv   = (const float*)d_in[2];
  const float* mu_w1  = (const float*)d_in[3];
  const float* rho_w1 = (const float*)d_in[4];
  const float* mu_b1  = (const float*)d_in[5];
  const float* rho_b1 = (const float*)d_in[6];
  const float* eps_w1 = (const float*)d_in[7];
  const float* eps_b1 = (const float*)d_in[8];
  const float* mu_w2  = (const float*)d_in[9];
  const float* rho_w2 = (const float*)d_in[10];
  const float* mu_b2  = (const float*)d_in[11];
  const float* rho_b2 = (const float*)d_in[12];
  const float* eps_w2 = (const float*)d_in[13];
  const float* eps_b2 = (const float*)d_in[14];
  const float* mu_w3  = (const float*)d_in[15];
  const float* rho_w3 = (const float*)d_in[16];
  const float* mu_b3  = (const float*)d_in[17];
  const float* rho_b3 = (const float*)d_in[18];
  const float* eps_w3 = (const float*)d_in[19];
  const float* eps_b3 = (const float*)d_in[20];

  const int B = in_sizes[0] / 32;  // 16384

  float* ws    = (float*)d_ws;
  float*  kl    = ws + 0;
  float*  bias1 = ws + 16;
  float*  bias2 = ws + 32;
  float*  bias3 = ws + 48;
  float2* EU    = (float2*)(ws + 64);             // 63 float2 (128 floats)
  float*  A1    = ws + 192;                       // 32*63*16 = 32256
  float*  A2    = A1 + 32 * MPTS * 16;            // 16*63*16 = 16128
  float*  A3    = A2 + 16 * MPTS * 16;            // 16128
  float2* E1    = (float2*)(A3 + 16 * MPTS * 16); // B*32 float2
  float2* E2    = E1 + (size_t)B * 32;            // B*16 float2
  float2* E3    = E2 + (size_t)B * 16;            // B*16 float2

  zero_kl_kernel<<<1, 1, 0, stream>>>(kl);

  build_eu_kernel<<<1, 64, 0, stream>>>(U, EU);
  exp_pack_kernel<<<(B * 32 + 255) / 256, 256, 0, stream>>>(x, E1, B * 32);

  build_A_kernel<<<(32 * MPTS * 16 + 255) / 256, 256, 0, stream>>>(
      Rinv, mu_w1, rho_w1, eps_w1, A1, 32, 16);
  build_A_kernel<<<(16 * MPTS * 16 + 255) / 256, 256, 0, stream>>>(
      Rinv, mu_w2, rho_w2, eps_w2, A2, 16, 16);
  build_A_kernel<<<(16 * MPTS * 16 + 255) / 256, 256, 0, stream>>>(
      Rinv, mu_w3, rho_w3, eps_w3, A3, 16, 1);

  build_bias_kernel<<<1, 16, 0, stream>>>(mu_b1, rho_b1, eps_b1, bias1, 32, 16);
  build_bias_kernel<<<1, 16, 0, stream>>>(mu_b2, rho_b2, eps_b2, bias2, 16, 16);
  build_bias_kernel<<<1, 16, 0, stream>>>(mu_b3, rho_b3, eps_b3, bias3, 16, 1);

  kl_accum_kernel<<<(32 * 16 * MPTS + 255) / 256, 256, 0, stream>>>(mu_w1, rho_w1, 32 * 16 * MPTS, kl);
  kl_accum_kernel<<<(32 * 16 + 255) / 256, 256, 0, stream>>>(mu_b1, rho_b1, 32 * 16, kl);
  kl_accum_kernel<<<(16 * 16 * MPTS + 255) / 256, 256, 0, stream>>>(mu_w2, rho_w2, 16 * 16 * MPTS, kl);
  kl_accum_kernel<<<(16 * 16 + 255) / 256, 256, 0, stream>>>(mu_b2, rho_b2, 16 * 16, kl);
  kl_accum_kernel<<<(16 * 1 * MPTS + 255) / 256, 256, 0, stream>>>(mu_w3, rho_w3, 16 * 1 * MPTS, kl);
  kl_accum_kernel<<<1, 256, 0, stream>>>(mu_b3, rho_b3, 16 * 1, kl);

  const int blocks = B / (16 * 8);  // 8 waves/block, one 16-row tile per wave
  tmgp_stage_wmma_kernel<32, 0><<<blocks, 256, 0, stream>>>(E1, EU, A1, bias1, E2, nullptr);
  tmgp_stage_wmma_kernel<16, 0><<<blocks, 256, 0, stream>>>(E2, EU, A2, bias2, E3, nullptr);
  tmgp_stage_wmma_kernel<16, 1><<<blocks, 256, 0, stream>>>(E3, EU, A3, bias3, nullptr, (float*)d_out);

  write_scalar_kernel<<<1, 1, 0, stream>>>(kl, (float*)d_out + B);
}